// Net_83966610636905
// MI455X (gfx1250) — compile-verified
//
#include <hip/hip_runtime.h>

// MI455X / gfx1250, wave32. fp32 WMMA path: V_WMMA_F32_16X16X4_F32.
// - Big weight fragment sets in LDS (pre-swizzled, conflict-free b64 loads),
//   wave stays under 256 VGPRs (no s_set_vgpr_msb churn).
// - All biases folded into the GEMMs via padded/extra K-steps.
// - ReLU as fmaxf (single v_max_num_f32), tanh via V_TANH_F32 when available.
typedef __attribute__((ext_vector_type(2))) float v2f;
typedef __attribute__((ext_vector_type(8))) float v8f;

#define STRIDE 104            // per-row LDS stride (floats), 8B-aligned rows
#define WAVES_PER_BLOCK 4
#define NTHREADS (WAVES_PER_BLOCK * 32)
#define BETA 3.0f
#define MAX_A 5.0f

static __device__ __forceinline__ v8f wmma4(v2f a, v2f b, v8f c) {
  // (neg_a, A, neg_b, B, c_mod, C, reuse_a, reuse_b)
  return __builtin_amdgcn_wmma_f32_16x16x4_f32(
      false, a, false, b, (short)0, c, false, false);
}

static __device__ __forceinline__ float fast_tanh(float x) {
#if __has_builtin(__builtin_amdgcn_tanhf)
  return __builtin_amdgcn_tanhf(x);          // V_TANH_F32 (1 TRANS op)
#elif __has_builtin(__builtin_amdgcn_tanh_f32)
  return __builtin_amdgcn_tanh_f32(x);
#else
  // tanh(x) = 1 - 2/(exp(2x)+1); saturates correctly via inf/0.
  float e = __expf(2.0f * x);
  return 1.0f - 2.0f / (e + 1.0f);
#endif
}

__global__ __launch_bounds__(NTHREADS)
void rbf_net_kernel(const float* __restrict__ s,  const float* __restrict__ a,
                    const float* __restrict__ W1, const float* __restrict__ b1,
                    const float* __restrict__ W2, const float* __restrict__ b2,
                    const float* __restrict__ L1, const float* __restrict__ bl1,
                    const float* __restrict__ WL2,const float* __restrict__ bL2,
                    float* __restrict__ out, int Btot, int nchunks) {
  // ---- LDS: per-wave transpose scratch + shared weight fragments ---------
  __shared__ float scratch[WAVES_PER_BLOCK * 16 * STRIDE];   // 26624 B
  __shared__ v2f   wfrag1[7 * 3 * 32];                       //  5376 B
  __shared__ v2f   lfrag1[7 * 3 * 32];                       //  5376 B
  __shared__ v2f   wlfrag[26 * 4 * 32];                      // 26624 B
                                                             // total 64000 B
  const int tid  = threadIdx.x;
  const int wid  = tid >> 5;
  const int lane = tid & 31;
  const int lid  = lane & 15;   // column / row-in-tile index
  const int kh   = lane >> 4;   // K half-select for fp32 A/B fragments
  float* hv = scratch + wid * (16 * STRIDE);

  // ---- cooperative fill of LDS weight fragments --------------------------
  // B-frag layout for 16x16x4 f32: lane = N column; VGPR0/1 = K{0,1}
  // (lanes 0-15) or K{2,3} (lanes 16-31). Frag kk covers K = 4*kk..4*kk+3.
  // Stage1/3: K-slot 10 carries the bias row (A supplies 1.0 there).
  for (int idx = tid; idx < 7 * 3 * 32; idx += NTHREADS) {
    int t = idx / 96, kk = (idx / 32) % 3, ln = idx & 31;
    int lidf = ln & 15, khf = ln >> 4;
    int col = t * 16 + lidf, k = kk * 4 + khf * 2;
    v2f w = {0.0f, 0.0f}, l = {0.0f, 0.0f};
    if (col < 100) {
      if (k < 10) {              // k even => k+1 <= 9 whenever k <= 8
        w.x = W1[k * 100 + col];  w.y = W1[(k + 1) * 100 + col];
        l.x = L1[k * 100 + col];  l.y = L1[(k + 1) * 100 + col];
      } else {                   // k == 10: bias row
        w.x = b1[col];
        l.x = bl1[col];
      }
    }
    wfrag1[idx] = w;
    lfrag1[idx] = l;
  }
  // Stage4: 26th K-step (K=100) carries the bL2 bias row.
  for (int idx = tid; idx < 26 * 4 * 32; idx += NTHREADS) {
    int kk = idx / 128, t = (idx / 32) & 3, ln = idx & 31;
    int lidf = ln & 15, khf = ln >> 4;
    int col = t * 16 + lidf, k = kk * 4 + khf * 2;   // col = 2*n + d
    v2f w = {0.0f, 0.0f};
    if (col < 60) {
      if (kk < 25) {
        int n = col >> 1, d = col & 1;
        int base = n * 200 + k * 2 + d;   // WL2[n][k][d]
        w.x = WL2[base]; w.y = WL2[base + 2];
      } else if (khf == 0) {              // K == 100: bias row
        w.x = bL2[col];
      }
    }
    wlfrag[idx] = w;
  }

  // ---- small weight set stays in VGPRs (W2 + bias K-step) ----------------
  v2f w2f[2][26];             // K = 100 (+ bias step), N tiles {0..15},{16..31}
#pragma unroll
  for (int t = 0; t < 2; ++t) {
    int col = t * 16 + lid;
#pragma unroll
    for (int kk = 0; kk < 25; ++kk) {
      int k = kk * 4 + kh * 2;
      v2f w = {0.0f, 0.0f};
      if (col < 30) { w.x = W2[k * 30 + col]; w.y = W2[(k + 1) * 30 + col]; }
      w2f[t][kk] = w;
    }
    v2f wb = {0.0f, 0.0f};
    if (col < 30 && kh == 0) wb.x = b2[col];   // K == 100: bias row
    w2f[t][25] = wb;
  }

  __syncthreads();   // weight fragments ready (read-only afterwards)

  // Constant-1 A fragment for the bias K-step (K=100 -> lanes 0-15, elt 0).
  const v2f hfb = kh ? (v2f){0.0f, 0.0f} : (v2f){1.0f, 0.0f};

  // ---------------- grid-stride loop over 16-row chunks --------------------
  const int gwave  = blockIdx.x * WAVES_PER_BLOCK + wid;
  const int nwaves = gridDim.x * WAVES_PER_BLOCK;

  for (int chunk = gwave; chunk < nchunks; chunk += nwaves) {
    const int row0 = chunk * 16;
    const int grow = row0 + lid;          // this lane's batch row (A frags)
    const long rc  = (grow < Btot) ? (long)grow : (long)(Btot - 1); // clamped

    // A fragments of s (shared by stage 1 and stage 3). 16x4 f32 A layout:
    // lane = M row; VGPR0/1 = K{0,1} (lanes 0-15) / K{2,3} (lanes 16-31).
    // Frag 2: lanes 16-31 carry {1,0} at K=10 to activate the bias row.
    const float* srow = s + rc * 10;
    v2f af0  = *(const v2f*)(srow + kh * 2);
    v2f af1  = *(const v2f*)(srow + 4 + kh * 2);
    v2f af2l = *(const v2f*)(srow + 8);            // unconditional, in-bounds
    v2f af2  = kh ? (v2f){1.0f, 0.0f} : af2l;      // select, no branch
    v2f af[3] = {af0, af1, af2};

    // -------- stage 1: hv = relu(s @ W1 + b1) -> LDS (bias in-GEMM) -------
#pragma unroll
    for (int t = 0; t < 7; ++t) {
      v8f acc = {0, 0, 0, 0, 0, 0, 0, 0};
#pragma unroll
      for (int kk = 0; kk < 3; ++kk)
        acc = wmma4(af[kk], wfrag1[(t * 3 + kk) * 32 + lane], acc);
      int col = t * 16 + lid;
      if (col < 100) {
#pragma unroll
        for (int j = 0; j < 8; ++j)
          hv[(j + 8 * kh) * STRIDE + col] = fmaxf(acc[j], 0.0f);
      }
    }

    // -------- stage 2: values = hv @ W2 + b2 (K = 100 + bias step) --------
    v8f vacc[2];
    vacc[0] = (v8f){0, 0, 0, 0, 0, 0, 0, 0};
    vacc[1] = (v8f){0, 0, 0, 0, 0, 0, 0, 0};
#pragma unroll
    for (int kk = 0; kk < 26; ++kk) {
      v2f hf = (kk < 25) ? *(const v2f*)&hv[lid * STRIDE + kk * 4 + kh * 2]
                         : hfb;
      vacc[0] = wmma4(hf, w2f[0][kk], vacc[0]);
      vacc[1] = wmma4(hf, w2f[1][kk], vacc[1]);
    }

    // -------- stage 3: hl = relu(s @ L1 + bl1) -> LDS (reuse buffer) ------
#pragma unroll
    for (int t = 0; t < 7; ++t) {
      v8f acc = {0, 0, 0, 0, 0, 0, 0, 0};
#pragma unroll
      for (int kk = 0; kk < 3; ++kk)
        acc = wmma4(af[kk], lfrag1[(t * 3 + kk) * 32 + lane], acc);
      int col = t * 16 + lid;
      if (col < 100) {
#pragma unroll
        for (int j = 0; j < 8; ++j)
          hv[(j + 8 * kh) * STRIDE + col] = fmaxf(acc[j], 0.0f);
      }
    }

    // -------- stage 4: cent_pre = hl @ WL2r + bL2 (K=100+bias, N=60) ------
    v8f cacc[4];
#pragma unroll
    for (int t = 0; t < 4; ++t) cacc[t] = (v8f){0, 0, 0, 0, 0, 0, 0, 0};
#pragma unroll
    for (int kk = 0; kk < 26; ++kk) {
      v2f hf = (kk < 25) ? *(const v2f*)&hv[lid * STRIDE + kk * 4 + kh * 2]
                         : hfb;
#pragma unroll
      for (int t = 0; t < 4; ++t)
        cacc[t] = wmma4(hf, wlfrag[(kk * 4 + t) * 32 + lane], cacc[t]);
    }

    // -------- epilogue staging: tanh in tile layout -> LDS ----------------
    // centroids at cols [0,60), values at cols [64,94)
#pragma unroll
    for (int t = 0; t < 4; ++t) {
      int col = t * 16 + lid;
      if (col < 60) {
#pragma unroll
        for (int j = 0; j < 8; ++j)
          hv[(j + 8 * kh) * STRIDE + col] = MAX_A * fast_tanh(cacc[t][j]);
      }
    }
#pragma unroll
    for (int t = 0; t < 2; ++t) {
      int n = t * 16 + lid;
      if (n < 30) {
#pragma unroll
        for (int j = 0; j < 8; ++j)
          hv[(j + 8 * kh) * STRIDE + 64 + n] = vacc[t][j];
      }
    }

    // -------- RBF softmax reduction: 16 rows x 2 halves of N=30 -----------
    {
      int row  = lane & 15;
      int half = lane >> 4;
      long r   = (long)row0 + row;
      long ra  = (r < Btot) ? r : (long)(Btot - 1);
      v2f av   = *(const v2f*)(a + ra * 2);
      float a0 = av.x, a1 = av.y;
      float ssum = 0.0f, wsum = 0.0f;
#pragma unroll
      for (int i = 0; i < 15; ++i) {
        int n    = half * 15 + i;
        float c0 = hv[row * STRIDE + 2 * n];
        float c1 = hv[row * STRIDE + 2 * n + 1];
        float v  = hv[row * STRIDE + 64 + n];
        float d0 = c0 - a0, d1 = c1 - a1;
        float e  = __expf(-BETA * sqrtf(d0 * d0 + d1 * d1));
        ssum += e;
        wsum += e * v;
      }
      ssum += __shfl_xor(ssum, 16, 32);
      wsum += __shfl_xor(wsum, 16, 32);
      if (half == 0 && r < Btot) out[r] = wsum / ssum;
    }
  }
}

extern "C" void kernel_launch(void* const* d_in, const int* in_sizes, int n_in,
                              void* d_out, int out_size, void* d_ws, size_t ws_size,
                              hipStream_t stream) {
  const float* s   = (const float*)d_in[0];
  const float* a   = (const float*)d_in[1];
  const float* W1  = (const float*)d_in[2];
  const float* b1  = (const float*)d_in[3];
  const float* W2  = (const float*)d_in[4];
  const float* b2  = (const float*)d_in[5];
  const float* L1  = (const float*)d_in[6];
  const float* bl1 = (const float*)d_in[7];
  const float* WL2 = (const float*)d_in[8];
  const float* bL2 = (const float*)d_in[9];
  float* out = (float*)d_out;

  int Btot    = in_sizes[0] / 10;
  int nchunks = (Btot + 15) / 16;
  int blocks  = (nchunks + WAVES_PER_BLOCK - 1) / WAVES_PER_BLOCK;
  if (blocks > 1024) blocks = 1024;
  if (blocks < 1) blocks = 1;

  rbf_net_kernel<<<blocks, NTHREADS, 0, stream>>>(
      s, a, W1, b1, W2, b2, L1, bl1, WL2, bL2, out, Btot, nchunks);
}